// EpiSIGNetV3_15204184227950
// MI455X (gfx1250) — compile-verified
//
#include <hip/hip_runtime.h>
#include <math.h>

// EpiSIGNetV3 on gfx1250 (MI455X). wave32; WMMA f32 16x16x4 for attention.
#define Bn   16
#define Tn   32
#define Nn   1024
#define FCc  16
#define HIDc 32
#define HORc 12
#define GHc  16
#define EPSf 1e-5f

typedef __attribute__((ext_vector_type(2))) float v2f;
typedef __attribute__((ext_vector_type(8))) float v8f;

__device__ __forceinline__ float sigm(float x){ return 1.f/(1.f+__expf(-x)); }
__device__ __forceinline__ float bnrelu(float x, float g, float b, float m, float v){
  float y = (x-m)*g*rsqrtf(v+EPSf)+b; return fmaxf(y,0.f);
}

struct DSP { const float *dww,*dwb,*b1g,*b1b,*b1m,*b1v,*pww,*pwb,*b2g,*b2b,*b2m,*b2v; };
struct EncArgs { const float* x; DSP feat, sc0, sc1, sc2;
                 const float *scale_w,*proj_w,*proj_b,*lng,*lnb; float* feats; };

// ---------------- Kernel 1: temporal encoder + proj + LN + relu -> feats[B*N,32]
// one wave (32 threads = T) per (b,n) sequence; all conv intermediates in LDS.
__global__ __launch_bounds__(32) void enc_kernel(EncArgs A) {
  const int s = blockIdx.x, b = s >> 10, n = s & 1023, t = threadIdx.x;
  __shared__ float xs[Tn];
  __shared__ float fbuf[FCc][Tn];
  __shared__ float ybuf[FCc][Tn];
  __shared__ float fused[FCc][Tn];
  xs[t] = A.x[(b*Tn + t)*Nn + n];
  __syncthreads();
  {
    float xm1 = (t>0)? xs[t-1] : 0.f;
    float xp1 = (t<Tn-1)? xs[t+1] : 0.f;
    float y = A.feat.dww[0]*xm1 + A.feat.dww[1]*xs[t] + A.feat.dww[2]*xp1 + A.feat.dwb[0];
    y = bnrelu(y, A.feat.b1g[0], A.feat.b1b[0], A.feat.b1m[0], A.feat.b1v[0]);
    #pragma unroll
    for (int c=0;c<FCc;c++){
      float z = A.feat.pww[c]*y + A.feat.pwb[c];
      fbuf[c][t]  = bnrelu(z, A.feat.b2g[c], A.feat.b2b[c], A.feat.b2m[c], A.feat.b2v[c]);
      fused[c][t] = 0.f;
    }
  }
  __syncthreads();
  float e0=A.scale_w[0], e1=A.scale_w[1], e2=A.scale_w[2];
  float mx = fmaxf(e0, fmaxf(e1,e2));
  e0=__expf(e0-mx); e1=__expf(e1-mx); e2=__expf(e2-mx);
  const float esum = e0+e1+e2;
  const float wsc[3] = { e0/esum, e1/esum, e2/esum };
  const DSP scs[3] = { A.sc0, A.sc1, A.sc2 };
  for (int i=0;i<3;i++) {
    const DSP S = scs[i];
    const int d = 1<<i;
    #pragma unroll
    for (int c=0;c<FCc;c++){
      float l  = (t-d>=0)? fbuf[c][t-d] : 0.f;
      float rr = (t+d<Tn)? fbuf[c][t+d] : 0.f;
      float y = S.dww[c*3+0]*l + S.dww[c*3+1]*fbuf[c][t] + S.dww[c*3+2]*rr + S.dwb[c];
      ybuf[c][t] = bnrelu(y, S.b1g[c],S.b1b[c],S.b1m[c],S.b1v[c]);
    }
    __syncthreads();
    #pragma unroll
    for (int co=0;co<FCc;co++){
      float acc = S.pwb[co];
      #pragma unroll
      for (int ci=0;ci<FCc;ci++) acc += S.pww[co*FCc+ci]*ybuf[ci][t];
      fused[co][t] += wsc[i]*bnrelu(acc, S.b2g[co],S.b2b[co],S.b2m[co],S.b2v[co]);
    }
    __syncthreads();
  }
  // projection: lane h computes feats[s][h] = relu(LN(fused . proj_w[h]))
  const int h = t;
  float acc = A.proj_b[h];
  const float* wrow = A.proj_w + h*(FCc*Tn);
  for (int c=0;c<FCc;c++)
    #pragma unroll
    for (int tt=0;tt<Tn;tt++) acc += fused[c][tt]*wrow[c*Tn+tt];
  float ssum = acc;
  #pragma unroll
  for (int m=16;m>=1;m>>=1) ssum += __shfl_xor(ssum, m, 32);
  const float mean = ssum*(1.f/32.f);
  const float dd = acc - mean;
  float vv = dd*dd;
  #pragma unroll
  for (int m=16;m>=1;m>>=1) vv += __shfl_xor(vv, m, 32);
  const float var = vv*(1.f/32.f);
  A.feats[s*HIDc + h] = fmaxf(dd*rsqrtf(var+EPSf)*A.lng[h] + A.lnb[h], 0.f);
}

// ---------------- Kernel 2: delay mixing + comb + qkv projection -> Q/K/V [B,H,N,8]
__global__ __launch_bounds__(96) void qkv_kernel(
    const float* __restrict__ x, const float* __restrict__ feats,
    const float* __restrict__ dlog, const float* __restrict__ dpw, const float* __restrict__ dpb,
    const float* __restrict__ qw, const float* __restrict__ qb,
    float* __restrict__ Qb, float* __restrict__ Kb, float* __restrict__ Vb) {
  const int s = blockIdx.x, b = s>>10, n = s&1023, j = threadIdx.x;
  __shared__ float comb[HIDc];
  if (j < HIDc) {
    float dl[8]; float mx = -3.0e38f;
    #pragma unroll
    for (int l=0;l<8;l++){ dl[l]=dlog[l]; mx=fmaxf(mx,dl[l]); }
    float se = 0.f;
    #pragma unroll
    for (int l=0;l<8;l++){ dl[l]=__expf(dl[l]-mx); se+=dl[l]; }
    float delayed = 0.f;
    #pragma unroll
    for (int l=0;l<8;l++) delayed += (dl[l]/se)*x[(b*Tn + (Tn-1-l))*Nn + n];
    comb[j] = feats[s*HIDc+j] + 0.2f*(delayed*dpw[j] + dpb[j]);
  }
  __syncthreads();
  float acc = qb[j];
  #pragma unroll
  for (int c=0;c<HIDc;c++) acc += qw[j*HIDc+c]*comb[c];
  const int which = j>>5, h = (j>>3)&3, d = j&7;
  float* dst = (which==0)? Qb : ((which==1)? Kb : Vb);
  dst[((b*4+h)*Nn + n)*8 + d] = acc;
}

// ---------------- Kernel 3: fused flash attention with V_WMMA_F32_16X16X4_F32
// one wave per (b, h, 16-query-row tile). Online softmax; P staged via 1KB LDS
// to convert C-layout -> A-layout between the QK^T and P@V GEMMs.
struct AttnArgs { const float *Q,*K,*V,*adj,*geo; float* att; };
__global__ __launch_bounds__(32) void attn_kernel(AttnArgs A) {
  const int nt = blockIdx.x, h = blockIdx.y, b = blockIdx.z;
  const int lane = threadIdx.x;
  const int r    = lane & 15;     // row (A/C M-index) or column (B N-index)
  const int khi  = lane >> 4;     // which K-pair this half-wave holds (f32 layout)
  const int n0   = nt*16;
  const float* Qp = A.Q + (size_t)((b*4+h)*Nn)*8;
  const float* Kp = A.K + (size_t)((b*4+h)*Nn)*8;
  const float* Vp = A.V + (size_t)((b*4+h)*Nn)*8;
  const float gb  = 1.f/(1.f+__expf(-A.geo[0]));
  const float sc0 = (1.f-gb)*0.35355339059f;   // (1-gb)/sqrt(8)
  const float ab  = gb*5.f;
  // A-fragments of Q tile (16x8, two K=4 steps): lanes<16 hold K={k0,k0+1}, lanes>=16 K={k0+2,k0+3}
  v2f qa0, qa1;
  qa0.x = Qp[(n0+r)*8 + 2*khi + 0];  qa0.y = Qp[(n0+r)*8 + 2*khi + 1];
  qa1.x = Qp[(n0+r)*8 + 4 + 2*khi];  qa1.y = Qp[(n0+r)*8 + 5 + 2*khi];
  v8f o = {};
  __shared__ float stile[16][17];
  __shared__ float rowM[16], rowL[16], rowA[16];
  if (lane < 16){ rowM[lane] = -3.0e38f; rowL[lane] = 0.f; }
  __syncthreads();
  for (int mt = 0; mt < Nn/16; ++mt) {
    const int m0 = mt*16;
    // S_tile = Q_tile(16x8) @ K_tile^T(8x16): two chained f32 WMMAs
    v8f c = {};
    v2f bb;
    bb.x = Kp[(m0+r)*8 + 2*khi + 0];  bb.y = Kp[(m0+r)*8 + 2*khi + 1];
    c = __builtin_amdgcn_wmma_f32_16x16x4_f32(false, qa0, false, bb, (short)0, c, false, false);
    bb.x = Kp[(m0+r)*8 + 4 + 2*khi];  bb.y = Kp[(m0+r)*8 + 5 + 2*khi];
    c = __builtin_amdgcn_wmma_f32_16x16x4_f32(false, qa1, false, bb, (short)0, c, false, false);
    // mix with adjacency prior; spill C-layout tile to LDS
    #pragma unroll
    for (int i=0;i<8;i++){
      const int ql = i + 8*khi;                    // C layout: M = i (+8 for upper half)
      stile[ql][r] = sc0*c[i] + ab*A.adj[(size_t)(n0+ql)*Nn + m0 + r];
    }
    __syncthreads();
    // online softmax bookkeeping, one row per lane (lanes 0..15)
    if (lane < 16) {
      float mOld = rowM[lane], mNew = mOld;
      #pragma unroll
      for (int jc=0;jc<16;jc++) mNew = fmaxf(mNew, stile[lane][jc]);
      const float alpha = __expf(mOld - mNew);
      float ss = 0.f;
      #pragma unroll
      for (int jc=0;jc<16;jc++){ float p=__expf(stile[lane][jc]-mNew); stile[lane][jc]=p; ss+=p; }
      rowL[lane] = rowL[lane]*alpha + ss;
      rowM[lane] = mNew;
      rowA[lane] = alpha;
    }
    __syncthreads();
    #pragma unroll
    for (int i=0;i<8;i++) o[i] *= rowA[i + 8*khi];
    // O += P(16x16) @ V_tile(16x8): four K=4 WMMA steps; P read back in A layout
    const float msk = (r < 8) ? 1.f : 0.f;        // pad V columns 8..15 with zeros
    const int cv = r & 7;
    #pragma unroll
    for (int kk=0;kk<4;kk++){
      v2f pa, vb;
      pa.x = stile[r][4*kk + 2*khi + 0];
      pa.y = stile[r][4*kk + 2*khi + 1];
      const int k0 = m0 + 4*kk + 2*khi;
      vb.x = Vp[(k0+0)*8 + cv]*msk;
      vb.y = Vp[(k0+1)*8 + cv]*msk;
      o = __builtin_amdgcn_wmma_f32_16x16x4_f32(false, pa, false, vb, (short)0, o, false, false);
    }
    __syncthreads();
  }
  if (r < 8) {
    #pragma unroll
    for (int i=0;i<8;i++){
      const int ql = i + 8*khi;
      A.att[(size_t)(b*Nn + n0 + ql)*HIDc + h*8 + r] = o[i] / rowL[ql];
    }
  }
}

// ---------------- Kernel 4: out projection + residual + LN -> feats2[B*N,32]
__global__ __launch_bounds__(32) void outln_kernel(
    const float* __restrict__ att, const float* __restrict__ ow, const float* __restrict__ ob,
    const float* __restrict__ feats, const float* __restrict__ ng, const float* __restrict__ nb,
    float* __restrict__ feats2) {
  const int s = blockIdx.x, h = threadIdx.x;
  __shared__ float arow[HIDc];
  arow[h] = att[s*HIDc+h];
  __syncthreads();
  float acc = ob[h] + feats[s*HIDc+h];
  #pragma unroll
  for (int c=0;c<HIDc;c++) acc += ow[h*HIDc+c]*arow[c];
  float ssum = acc;
  #pragma unroll
  for (int m=16;m>=1;m>>=1) ssum += __shfl_xor(ssum, m, 32);
  const float mean = ssum*(1.f/32.f);
  const float dd = acc - mean;
  float vv = dd*dd;
  #pragma unroll
  for (int m=16;m>=1;m>>=1) vv += __shfl_xor(vv, m, 32);
  feats2[s*HIDc+h] = dd*rsqrtf(vv*(1.f/32.f)+EPSf)*ng[h] + nb[h];
}

// ---------------- Kernel 5: GRU rollout + gate + decay + highway -> out[B,N,12]
struct HeadArgs {
  const float *x,*feats2,*wih,*whh,*bih,*bhh,*hidw,*hidb,*op1w,*op1b,*op2w,*op2b,
              *ld,*rg1w,*rg1b,*rg2w,*rg2b,*hww,*hwb;
  float* out;
};
__global__ __launch_bounds__(256) void head_kernel(HeadArgs A) {
  __shared__ float swhh[3*GHc*GHc];
  for (int idx = threadIdx.x; idx < 3*GHc*GHc; idx += blockDim.x) swhh[idx] = A.whh[idx];
  __syncthreads();
  const int s = blockIdx.x*blockDim.x + threadIdx.x;
  const int b = s>>10, n = s&1023;
  float fr[HIDc];
  #pragma unroll
  for (int c=0;c<HIDc;c++) fr[c] = A.feats2[s*HIDc+c];
  float hst[GHc];
  #pragma unroll
  for (int i=0;i<GHc;i++){
    float a = A.hidb[i];
    #pragma unroll
    for (int c=0;c<HIDc;c++) a += A.hidw[i*HIDc+c]*fr[c];
    hst[i] = a;
  }
  float t1[8];
  #pragma unroll
  for (int i=0;i<8;i++){
    float a = A.rg1b[i];
    #pragma unroll
    for (int c=0;c<HIDc;c++) a += A.rg1w[i*HIDc+c]*fr[c];
    t1[i] = fmaxf(a, 0.f);
  }
  float gate[HORc];
  #pragma unroll
  for (int i=0;i<HORc;i++){
    float a = A.rg2b[i];
    #pragma unroll
    for (int k=0;k<8;k++) a += A.rg2w[i*8+k]*t1[k];
    gate[i] = sigm(a);
  }
  const float last = A.x[(b*Tn + Tn-1)*Nn + n];
  const float dec  = __expf(A.ld[0]);
  float xin = last;
  float res[HORc];
  for (int st=0; st<HORc; ++st) {
    float hnew[GHc];
    #pragma unroll
    for (int i=0;i<GHc;i++){
      float gr = A.wih[i]*xin + A.bih[i];
      float gz = A.wih[GHc+i]*xin + A.bih[GHc+i];
      float gn = A.wih[2*GHc+i]*xin + A.bih[2*GHc+i];
      float hr = A.bhh[i], hz = A.bhh[GHc+i], hn = A.bhh[2*GHc+i];
      #pragma unroll
      for (int k=0;k<GHc;k++){
        hr += swhh[(i)*GHc+k]*hst[k];
        hz += swhh[(GHc+i)*GHc+k]*hst[k];
        hn += swhh[(2*GHc+i)*GHc+k]*hst[k];
      }
      const float r = sigm(gr+hr), z = sigm(gz+hz);
      const float nn = tanhf(gn + r*hn);
      hnew[i] = (1.f-z)*nn + z*hst[i];
    }
    #pragma unroll
    for (int i=0;i<GHc;i++) hst[i] = hnew[i];
    float tt[8];
    #pragma unroll
    for (int i=0;i<8;i++){
      float a = A.op1b[i];
      #pragma unroll
      for (int k=0;k<GHc;k++) a += A.op1w[i*GHc+k]*hst[k];
      tt[i] = fmaxf(a, 0.f);
    }
    float pred = A.op2b[0];
    #pragma unroll
    for (int k=0;k<8;k++) pred += A.op2w[k]*tt[k];
    xin = pred;
    const float dpred = last * __expf(-dec*(float)(st+1));
    res[st] = gate[st]*pred + (1.f-gate[st])*dpred;
  }
  #pragma unroll
  for (int ho=0;ho<HORc;ho++){
    float hw = A.hwb[ho];
    #pragma unroll
    for (int tq=0;tq<4;tq++) hw += A.x[(b*Tn + (Tn-4)+tq)*Nn + n]*A.hww[ho*4+tq];
    A.out[s*HORc + ho] = res[ho] + hw;
  }
}

// ---------------- host launcher --------------------------------------------
extern "C" void kernel_launch(void* const* d_in, const int* in_sizes, int n_in,
                              void* d_out, int out_size, void* d_ws, size_t ws_size,
                              hipStream_t stream) {
  (void)in_sizes; (void)n_in; (void)out_size; (void)ws_size;
  auto F = [&](int i){ return (const float*)d_in[i]; };
  auto mkds = [&](int base){
    DSP r; r.dww=F(base); r.dwb=F(base+1);
    r.b1g=F(base+2); r.b1b=F(base+3); r.b1m=F(base+4); r.b1v=F(base+5);
    r.pww=F(base+6); r.pwb=F(base+7);
    r.b2g=F(base+8); r.b2b=F(base+9); r.b2m=F(base+10); r.b2v=F(base+11);
    return r;
  };
  // workspace layout (floats)
  float* ws     = (float*)d_ws;
  float* feats  = ws;                 // 16384*32
  float* Qb     = ws +  524288;       // [B,H,N,8]
  float* Kb     = ws + 1048576;
  float* Vb     = ws + 1572864;
  float* attb   = ws + 2097152;       // 16384*32
  float* feats2 = ws + 2621440;       // 16384*32

  EncArgs ea;
  ea.x = F(0);
  ea.feat = mkds(1); ea.sc0 = mkds(13); ea.sc1 = mkds(25); ea.sc2 = mkds(37);
  ea.scale_w = F(49); ea.proj_w = F(50); ea.proj_b = F(51);
  ea.lng = F(52); ea.lnb = F(53); ea.feats = feats;
  enc_kernel<<<Bn*Nn, 32, 0, stream>>>(ea);

  qkv_kernel<<<Bn*Nn, 96, 0, stream>>>(F(0), feats, F(54), F(61), F(62),
                                       F(55), F(56), Qb, Kb, Vb);

  AttnArgs aa; aa.Q = Qb; aa.K = Kb; aa.V = Vb; aa.adj = F(63); aa.geo = F(64); aa.att = attb;
  attn_kernel<<<dim3(Nn/16, 4, Bn), 32, 0, stream>>>(aa);

  outln_kernel<<<Bn*Nn, 32, 0, stream>>>(attb, F(57), F(58), feats, F(59), F(60), feats2);

  HeadArgs ha;
  ha.x = F(0); ha.feats2 = feats2;
  ha.wih = F(65); ha.whh = F(66); ha.bih = F(67); ha.bhh = F(68);
  ha.hidw = F(69); ha.hidb = F(70); ha.op1w = F(71); ha.op1b = F(72);
  ha.op2w = F(73); ha.op2b = F(74); ha.ld = F(75);
  ha.rg1w = F(76); ha.rg1b = F(77); ha.rg2w = F(78); ha.rg2b = F(79);
  ha.hww = F(80); ha.hwb = F(81);
  ha.out = (float*)d_out;
  head_kernel<<<(Bn*Nn)/256, 256, 0, stream>>>(ha);
}